// ExpWCELoss_17849884082956
// MI455X (gfx1250) — compile-verified
//
#include <hip/hip_runtime.h>
#include <hip/hip_bf16.h>
#include <math.h>

// ExpWCELoss for B=32, C=4, H=512, W=512 (f32 in, f32 scalar out).
// Memory-bound streaming reduction: ~268 MB read -> ~11.5 us floor @ 23.3 TB/s.

namespace {

constexpr int   kCdim   = 4;
constexpr int   kHW     = 512 * 512;      // 262144
constexpr int   kHW4    = kHW / 4;        // 65536 (float4 chunks per (b,c) plane)
constexpr int   kNVox   = 32 * kHW;       // 8388608 voxels (B*H*W)
constexpr int   kNChunk = kNVox / 4;      // 2097152 float4 chunks over (B,H,W)
constexpr float kEps    = 1e-10f;

constexpr int kBlock = 256;               // 8 wave32 per block
constexpr int kGrid  = 1024;              // 262,144 threads -> exactly 8 chunks/thread, no tail

typedef __attribute__((ext_vector_type(2))) float v2f;
typedef __attribute__((ext_vector_type(4))) float v4f;
typedef __attribute__((ext_vector_type(8))) float v8f;

// Wave32 sum with a CDNA5 WMMA finish.
// shfl_xor(1,2,4,8): lanes 0-15 all hold L = sum(lanes 0..15), lanes 16-31 hold H.
// V_WMMA_F32_16X16X4_F32 with A VGPR0 = v, VGPR1 = 0 => row m of A = [L,0,H,0]
// (documented A layout); B = all-ones in both VGPRs (layout-independent).
// => D[m,n] = L + H = full wave sum, broadcast into every lane of D.
__device__ __forceinline__ float wave_sum32(float v) {
  v += __shfl_xor(v, 1, 32);
  v += __shfl_xor(v, 2, 32);
  v += __shfl_xor(v, 4, 32);
  v += __shfl_xor(v, 8, 32);
  v2f a; a.x = v;    a.y = 0.0f;
  v2f b; b.x = 1.0f; b.y = 1.0f;
  v8f c = {};
  v8f d = __builtin_amdgcn_wmma_f32_16x16x4_f32(
      /*neg_a=*/false, a, /*neg_b=*/false, b,
      /*c_mod=*/(short)0, c, /*reuse_a=*/false, /*reuse_b=*/false);
  return d[0];
}

__global__ void zero_ws_kernel(float* __restrict__ ws) {
  if (threadIdx.x < 8) ws[threadIdx.x] = 0.0f;
}

// kAtomicOut = true : atomically accumulate the 8 block partials into ws[0..7].
// kAtomicOut = false: write block partials to ws[blockIdx*8 + j] (contention-free).
template <bool kAtomicOut>
__global__ __launch_bounds__(kBlock) void wce_partial_kernel(
    const float* __restrict__ predict, const float* __restrict__ target,
    float* __restrict__ ws) {
  const v4f* __restrict__ Pv = (const v4f*)predict;
  const v4f* __restrict__ Tv = (const v4f*)target;

  float ce0 = 0.f, ce1 = 0.f, ce2 = 0.f, ce3 = 0.f;  // per-class CE partial sums
  float cn0 = 0.f, cn1 = 0.f, cn2 = 0.f, cn3 = 0.f;  // per-class argmax counts

  const int stride = kGrid * kBlock;
  for (int i = (int)(blockIdx.x * kBlock + threadIdx.x); i < kNChunk; i += stride) {
    const int b    = i >> 16;            // i / kHW4
    const int s4   = i & (kHW4 - 1);     // float4 offset inside the (H,W) plane
    const int base = b * (kCdim * kHW4) + s4;

    // 8 coalesced global_load_b128 streams; NT hint (no reuse, footprint > 192MB L2)
    const v4f t0 = __builtin_nontemporal_load(&Tv[base]);
    const v4f t1 = __builtin_nontemporal_load(&Tv[base + kHW4]);
    const v4f t2 = __builtin_nontemporal_load(&Tv[base + 2 * kHW4]);
    const v4f t3 = __builtin_nontemporal_load(&Tv[base + 3 * kHW4]);
    const v4f p0 = __builtin_nontemporal_load(&Pv[base]);
    const v4f p1 = __builtin_nontemporal_load(&Pv[base + kHW4]);
    const v4f p2 = __builtin_nontemporal_load(&Pv[base + 2 * kHW4]);
    const v4f p3 = __builtin_nontemporal_load(&Pv[base + 3 * kHW4]);

#pragma unroll
    for (int e = 0; e < 4; ++e) {
      const float a0 = t0[e], a1 = t1[e], a2 = t2[e], a3 = t3[e];
      // first-max argmax over C (matches jnp.argmax tie-breaking)
      int am = 0; float best = a0;
      if (a1 > best) { best = a1; am = 1; }
      if (a2 > best) { best = a2; am = 2; }
      if (a3 > best) { best = a3; am = 3; }
      // target is one-hot: sum_c -t_c*log(p_c+eps) == -t_am*log(p_am+eps)
      const float psel = (am == 0) ? p0[e] : (am == 1) ? p1[e]
                       : (am == 2) ? p2[e] : p3[e];
      const float v = -best * logf(psel + kEps);
      ce0 += (am == 0) ? v : 0.f;
      ce1 += (am == 1) ? v : 0.f;
      ce2 += (am == 2) ? v : 0.f;
      ce3 += (am == 3) ? v : 0.f;
      cn0 += (am == 0) ? 1.f : 0.f;
      cn1 += (am == 1) ? 1.f : 0.f;
      cn2 += (am == 2) ? 1.f : 0.f;
      cn3 += (am == 3) ? 1.f : 0.f;
    }
  }

  // Uniform control flow here (no tail) => EXEC all ones, legal for WMMA.
  float s[8];
  s[0] = wave_sum32(ce0); s[1] = wave_sum32(ce1);
  s[2] = wave_sum32(ce2); s[3] = wave_sum32(ce3);
  s[4] = wave_sum32(cn0); s[5] = wave_sum32(cn1);
  s[6] = wave_sum32(cn2); s[7] = wave_sum32(cn3);

  __shared__ float red[kBlock / 32][8];
  const int wave = threadIdx.x >> 5;
  const int lane = threadIdx.x & 31;
  if (lane == 0) {
#pragma unroll
    for (int j = 0; j < 8; ++j) red[wave][j] = s[j];
  }
  __syncthreads();
  if (threadIdx.x < 8) {
    float acc = 0.f;
#pragma unroll
    for (int w = 0; w < kBlock / 32; ++w) acc += red[w][threadIdx.x];
    if (kAtomicOut) {
      atomicAdd(&ws[threadIdx.x], acc);            // ws[0..3]=CE, ws[4..7]=counts
    } else {
      ws[blockIdx.x * 8 + threadIdx.x] = acc;      // per-block slot, no contention
    }
  }
}

__device__ __forceinline__ float final_combine(const float* s8) {
  const float voxels = (float)kNVox;
  float acc = 0.f;
#pragma unroll
  for (int c = 0; c < kCdim; ++c) {
    const float ce = s8[c] / voxels;               // mean over (B,H,W)
    const float w  = sqrtf(voxels / s8[4 + c]);    // sqrt(N / count_c)
    acc += ce * w;
  }
  return 0.25f * acc;                              // mean over C
}

// Second stage for the contention-free path: reduce kGrid x 8 partials.
// Wave j reduces accumulator j across all blocks, finishing with the WMMA wave-sum.
__global__ __launch_bounds__(256) void reduce_final_kernel(
    const float* __restrict__ ws, float* __restrict__ out) {
  const int wave = threadIdx.x >> 5;   // 0..7 -> accumulator index
  const int lane = threadIdx.x & 31;
  float acc = 0.f;
  for (int blk = lane; blk < kGrid; blk += 32)     // uniform trip count (32 iters)
    acc += ws[blk * 8 + wave];
  acc = wave_sum32(acc);                           // EXEC all ones here

  __shared__ float red[8];
  if (lane == 0) red[wave] = acc;
  __syncthreads();
  if (threadIdx.x == 0) out[0] = final_combine(red);
}

// Finalize for the atomic fallback path: ws[0..7] already hold global sums.
__global__ void finalize_small_kernel(const float* __restrict__ ws,
                                      float* __restrict__ out) {
  if (threadIdx.x == 0) {
    float s8[8];
#pragma unroll
    for (int j = 0; j < 8; ++j) s8[j] = ws[j];
    out[0] = final_combine(s8);
  }
}

}  // namespace

extern "C" void kernel_launch(void* const* d_in, const int* in_sizes, int n_in,
                              void* d_out, int out_size, void* d_ws, size_t ws_size,
                              hipStream_t stream) {
  (void)in_sizes; (void)n_in; (void)out_size;
  const float* predict = (const float*)d_in[0];
  const float* target  = (const float*)d_in[1];
  float* ws  = (float*)d_ws;
  float* out = (float*)d_out;

  constexpr size_t kWsNeeded = (size_t)kGrid * 8 * sizeof(float);  // 32 KB
  if (ws_size >= kWsNeeded) {
    // Contention-free two-phase reduction (deterministic, no atomics).
    wce_partial_kernel<false><<<kGrid, kBlock, 0, stream>>>(predict, target, ws);
    reduce_final_kernel<<<1, 256, 0, stream>>>(ws, out);
  } else {
    // Fallback: 8-float atomic accumulation in ws[0..7].
    zero_ws_kernel<<<1, 32, 0, stream>>>(ws);
    wce_partial_kernel<true><<<kGrid, kBlock, 0, stream>>>(predict, target, ws);
    finalize_small_kernel<<<1, 32, 0, stream>>>(ws, out);
  }
}